// WeightedEdgeConv_25924422598774
// MI455X (gfx1250) — compile-verified
//
#include <hip/hip_runtime.h>

// WeightedEdgeConv aggregate:  out[j] += ew[e] * x[i]   for each edge e=(i,j)
//
// x [50000, 64] f32, g [2, 1M] int32 (src row, then dst row), ew [1M] f32,
// out [50000, 64] f32.
//
// Phase A (per 16-edge block):
//   - wave 0: TDM gather -- one tensor_load_to_lds with a gather-mode D#
//     (16 x 16-bit row indices, tile_dim0=64 f32) pulls the 16 source rows
//     (4 KB) from L2-resident x into LDS. Indices come in via wide uniform
//     SMEM loads (full-block fast path) and SALU 16-bit packing.
//   - wave 1 (concurrently): stages the block's dst[] and ew[] into LDS so
//     Phase B never re-loads them per lane.
// Phase B: 16 lanes per edge read the LDS row (ds_load_b128), scale by ew[e],
//   scatter with no-return global_atomic_add_f32 -- the real bottleneck:
//   64M L2-side f32 RMWs; x (12.8 MB) and out (12.8 MB) both live in the
//   192 MB L2, so HBM traffic is only the ~12 MB edge stream.

#define C_FEAT 64
#define THREADS 256            // 8 wave32 per block
#define EDGES_PER_BLOCK 16
#define LANES_PER_EDGE 16
#define PREFETCH_EDGES 4096

typedef unsigned int v4u __attribute__((ext_vector_type(4)));
typedef int          v4i __attribute__((ext_vector_type(4)));
typedef int          v8i __attribute__((ext_vector_type(8)));

__global__ __launch_bounds__(THREADS) void weconv_tdm_scatter_kernel(
    const float* __restrict__ x,
    const int*   __restrict__ src,   // g[0][*]
    const int*   __restrict__ dst,   // g[1][*]
    const float* __restrict__ ew,
    float*       __restrict__ out,
    int n_edges, int n_nodes) {

    __shared__ __align__(16) float tile[EDGES_PER_BLOCK][C_FEAT];
    __shared__ int   sdst[EDGES_PER_BLOCK];
    __shared__ float sew [EDGES_PER_BLOCK];

    const int e0 = blockIdx.x * EDGES_PER_BLOCK;
    const int t  = threadIdx.x;

    // ---- Phase A (wave 1): stage dst/ew for the block into LDS.
    if (t >= 32 && t < 48) {
        int e = e0 + (t - 32);
        e = (e < n_edges) ? e : (n_edges - 1);
        sdst[t - 32] = dst[e];
    } else if (t >= 48 && t < 64) {
        int e = e0 + (t - 48);
        e = (e < n_edges) ? e : (n_edges - 1);
        sew[t - 48] = ew[e];
    }

    // ---- Phase A (wave 0): issue the TDM gather of 16 source rows into LDS.
    if (t < 32) {
        const int rem    = n_edges - e0;
        const int nvalid = (rem < EDGES_PER_BLOCK) ? rem : EDGES_PER_BLOCK;

        unsigned pk[8];
        if (rem >= EDGES_PER_BLOCK) {
            // Full block: straight consecutive uniform loads -> wide s_load.
            const int* sp = src + e0;
#pragma unroll
            for (int k = 0; k < 8; ++k) {
                const unsigned ia = (unsigned)sp[2 * k]     & 0xFFFFu;
                const unsigned ib = (unsigned)sp[2 * k + 1] & 0xFFFFu;
                pk[k] = ia | (ib << 16);
            }
        } else {
            // Tail block: clamp each edge index.
#pragma unroll
            for (int k = 0; k < 8; ++k) {
                int ea = e0 + 2 * k;
                int eb = e0 + 2 * k + 1;
                ea = (ea < n_edges) ? ea : (n_edges - 1);
                eb = (eb < n_edges) ? eb : (n_edges - 1);
                const unsigned ia = (unsigned)src[ea] & 0xFFFFu;
                const unsigned ib = (unsigned)src[eb] & 0xFFFFu;
                pk[k] = ia | (ib << 16);
            }
        }

        const unsigned lds_base =
            (unsigned)(unsigned long long)(&tile[0][0]);   // addr[31:0] == LDS offset
        const unsigned long long xa = (unsigned long long)x;

        // D# group 0: count=1, gather_mode=1 (16-bit indices), lds_addr,
        //             global_addr[56:0], type=2.
        v4u g0;
        g0[0] = 0x80000001u;                       // gather_mode<<31 | count=1
        g0[1] = lds_base;
        g0[2] = (unsigned)(xa & 0xFFFFFFFFu);
        g0[3] = (unsigned)((xa >> 32) & 0x1FFFFFFu) | (2u << 30);  // type=2

        // D# group 1: data_size=4B, tensor_dim0=64, tensor_dim1=n_nodes,
        //             tile_dim0=64, tile_dim1=nvalid, tensor_dim0_stride=64.
        v8i g1;
        g1[0] = (int)(2u << 16);                   // data_size = 2 (4 bytes)
        g1[1] = (int)((unsigned)C_FEAT << 16);     // tensor_dim0[15:0]
        g1[2] = (int)(((unsigned)n_nodes & 0xFFFFu) << 16);  // dim0 hi=0 | dim1 lo
        g1[3] = (int)((((unsigned)n_nodes >> 16) & 0xFFFFu)  // dim1 hi
                      | ((unsigned)C_FEAT << 16));           // tile_dim0
        g1[4] = nvalid;                            // tile_dim1 = #valid indices
        g1[5] = C_FEAT;                            // tensor_dim0_stride[31:0]
        g1[6] = 0;                                 // stride hi | dim1_stride (ignored)
        g1[7] = 0;

        v4i g2; g2[0] = (int)pk[0]; g2[1] = (int)pk[1]; g2[2] = (int)pk[2]; g2[3] = (int)pk[3];
        v4i g3; g3[0] = (int)pk[4]; g3[1] = (int)pk[5]; g3[2] = (int)pk[6]; g3[3] = (int)pk[7];

        // clang-23 toolchain: 6-arg builtin; extra int32x8 group zeroed (the
        // architectural D# is fully described by g0..g3).
        v8i gx; gx[0] = 0; gx[1] = 0; gx[2] = 0; gx[3] = 0;
                gx[4] = 0; gx[5] = 0; gx[6] = 0; gx[7] = 0;

        __builtin_amdgcn_tensor_load_to_lds(g0, g1, g2, g3, gx, /*cpol=*/0);
        __builtin_amdgcn_s_wait_tensorcnt(0);
    }
    __syncthreads();

    // ---- Phase B: scale + atomic scatter (16 lanes per edge, float4/lane).
    const int el = t >> 4;
    const int c  = (t & (LANES_PER_EDGE - 1)) << 2;
    const int e  = e0 + el;
    if (e >= n_edges) return;

    if (((t & (LANES_PER_EDGE - 1)) == 0) && (e + PREFETCH_EDGES < n_edges)) {
        __builtin_prefetch(src + e + PREFETCH_EDGES, 0, 0);
        __builtin_prefetch(dst + e + PREFETCH_EDGES, 0, 0);
        __builtin_prefetch(ew  + e + PREFETCH_EDGES, 0, 0);
    }

    const int   j = sdst[el];
    const float w = sew[el];

    const float4 xv = *reinterpret_cast<const float4*>(&tile[el][c]);  // ds_load_b128

    float* o = out + (long long)j * C_FEAT + c;
    (void)unsafeAtomicAdd(o + 0, xv.x * w);
    (void)unsafeAtomicAdd(o + 1, xv.y * w);
    (void)unsafeAtomicAdd(o + 2, xv.z * w);
    (void)unsafeAtomicAdd(o + 3, xv.w * w);
}

__global__ __launch_bounds__(256) void zero_out_kernel(float4* __restrict__ out,
                                                       int n4) {
    int idx = blockIdx.x * blockDim.x + threadIdx.x;
    if (idx < n4) {
        out[idx] = make_float4(0.0f, 0.0f, 0.0f, 0.0f);
    }
}

extern "C" void kernel_launch(void* const* d_in, const int* in_sizes, int n_in,
                              void* d_out, int out_size, void* d_ws, size_t ws_size,
                              hipStream_t stream) {
    const float* x  = (const float*)d_in[0];
    const int*   g  = (const int*)d_in[1];   // [2, E] flattened
    const float* ew = (const float*)d_in[2];

    const int n_edges = in_sizes[2];                 // E
    const int n_nodes = in_sizes[0] / C_FEAT;        // N
    const int* src = g;                              // g[0][*]
    const int* dst = g + n_edges;                    // g[1][*]

    float* out = (float*)d_out;

    // 1) Zero the (poisoned) output accumulator.
    int n4 = out_size >> 2;
    zero_out_kernel<<<(n4 + 255) / 256, 256, 0, stream>>>((float4*)out, n4);

    // 2) TDM-gather + atomic scatter, 16 edges per block.
    int blocks = (n_edges + EDGES_PER_BLOCK - 1) / EDGES_PER_BLOCK;
    weconv_tdm_scatter_kernel<<<blocks, THREADS, 0, stream>>>(x, src, dst, ew,
                                                              out, n_edges,
                                                              n_nodes);
}